// AtomwiseReduce_49976239456290
// MI455X (gfx1250) — compile-verified
//
#include <hip/hip_runtime.h>
#include <hip/hip_bf16.h>
#include <math.h>

#define N_ATOMS 400000
#define FEAT    512
#define NSEG    50000

typedef float v2f __attribute__((ext_vector_type(2)));
typedef float v4f __attribute__((ext_vector_type(4)));
typedef float v8f __attribute__((ext_vector_type(8)));

// first index i with arr[i] >= v (arr sorted ascending)
__device__ __forceinline__ int lower_bound_i32(const int* __restrict__ arr, int n, int v) {
    int lo = 0, hi = n;
    while (lo < hi) {
        int mid = (lo + hi) >> 1;
        if (arr[mid] < v) lo = mid + 1; else hi = mid;
    }
    return lo;
}

// float atomic max via integer atomics (order-isomorphic encoding), always available
__device__ __forceinline__ void atomicMaxF32(float* addr, float val) {
    if (val >= 0.0f) atomicMax((int*)addr, __float_as_int(val));
    else             atomicMin((unsigned int*)addr, __float_as_uint(val));
}

// ---------------------------------------------------------------------------
// Kernel 0: w_q[d] = sum_e W[e][d] * q[e]   (bias b cancels in softmax)
// ---------------------------------------------------------------------------
__global__ void wq_kernel(const float* __restrict__ W, const float* __restrict__ q,
                          float* __restrict__ wq) {
    __shared__ float s_q[FEAT];
    int tid = threadIdx.x;
    s_q[tid]       = q[tid];
    s_q[tid + 256] = q[tid + 256];
    __syncthreads();
    int d = blockIdx.x * 256 + tid;
    float acc = 0.0f;
    for (int e = 0; e < FEAT; ++e)
        acc += W[(size_t)e * FEAT + d] * s_q[e];
    wq[d] = acc;
}

// ---------------------------------------------------------------------------
// Kernel 1: init softmax stats (L2-resident, 0.4 MB)
// ---------------------------------------------------------------------------
__global__ void init_kernel(float* __restrict__ seg_max, float* __restrict__ denom) {
    int i = blockIdx.x * blockDim.x + threadIdx.x;
    if (i < NSEG) {
        seg_max[i] = __int_as_float(0xFF800000); // -inf
        denom[i]   = 0.0f;
    }
}

// ---------------------------------------------------------------------------
// Kernel 2: scores[n] = F[n,:] . w_q ; atomic segment max.
// One wave per row, NT streaming loads of features, wave32 shuffle reduce.
// ---------------------------------------------------------------------------
__global__ void scores_kernel(const float* __restrict__ feat,
                              const int*   __restrict__ ridx,
                              const float* __restrict__ wq,
                              float* __restrict__ scores,
                              float* __restrict__ seg_max) {
    __shared__ float s_wq[FEAT];
    int tid = threadIdx.x;
    s_wq[tid]       = wq[tid];
    s_wq[tid + 256] = wq[tid + 256];
    __syncthreads();

    int wave = tid >> 5;
    int lane = tid & 31;
    int n = blockIdx.x * 8 + wave;              // N_ATOMS = 50000 * 8 exactly
    const float* row = feat + (size_t)n * FEAT;

    // stream-ahead hint: rows of the next block (emits global_prefetch_b8)
    __builtin_prefetch(row + 8 * FEAT, 0, 0);

    float acc = 0.0f;
#pragma unroll
    for (int j = 0; j < 4; ++j) {
        int o = j * 128 + lane * 4;             // 16B-aligned, fully coalesced
        v4f f = __builtin_nontemporal_load((const v4f*)(row + o));
        acc += f.x * s_wq[o] + f.y * s_wq[o + 1] + f.z * s_wq[o + 2] + f.w * s_wq[o + 3];
    }
#pragma unroll
    for (int off = 16; off > 0; off >>= 1)
        acc += __shfl_down(acc, off, 32);

    if (lane == 0) {
        scores[n] = acc;
        atomicMaxF32(&seg_max[ridx[n]], acc);
    }
}

// ---------------------------------------------------------------------------
// Kernel 3: exp_s[n] = exp(scores[n] - seg_max[seg]); denom[seg] += exp_s[n]
// ---------------------------------------------------------------------------
__global__ void expsum_kernel(const float* __restrict__ scores,
                              const int*   __restrict__ ridx,
                              const float* __restrict__ seg_max,
                              float* __restrict__ exp_s,
                              float* __restrict__ denom) {
    int i = blockIdx.x * blockDim.x + threadIdx.x;
    if (i >= N_ATOMS) return;
    int s = ridx[i];
    float e = __expf(scores[i] - seg_max[s]);
    exp_s[i] = e;
    atomicAdd(&denom[s], e);
}

// ---------------------------------------------------------------------------
// Kernel 3.5: attn[n] = exp_s[n] / denom[seg]  (exact IEEE div, hoisted out of
// the WMMA loop) + build sentinel-padded residue index so the WMMA inner loop
// needs NO validity booleans: entries [N, N+4) get residue NSEG, which can
// never match a selector row (s0 + lm <= NSEG - 1).
// ---------------------------------------------------------------------------
__global__ void attn_kernel(const float* __restrict__ exp_s,
                            const int*   __restrict__ ridx,
                            const float* __restrict__ denom,
                            float* __restrict__ attn,
                            int*   __restrict__ ridx_pad) {
    int i = blockIdx.x * blockDim.x + threadIdx.x;
    if (i >= N_ATOMS + 4) return;
    if (i < N_ATOMS) {
        ridx_pad[i] = ridx[i];
        attn[i] = exp_s[i] / denom[ridx[i]];
    } else {
        ridx_pad[i] = NSEG;                    // sentinel: matches no block row
    }
}

// ---------------------------------------------------------------------------
// Kernel 4: segmented weighted sum as a scatter-GEMM on V_WMMA_F32_16X16X4_F32.
//   Workgroup = 16 consecutive segments; 8 waves x 4 D-tiles cover 512 features.
//   K-chunk = 4 consecutive atoms: A[m][k] = attn(atom k) iff atom k is in
//   segment m (weighted selector), B[k][n] = features[atom k][feature n].
//   Sorted residue_index -> uniform [start,end) via binary search, so EXEC
//   stays all-ones around the WMMA as the ISA requires. Tail atoms are masked
//   purely by the residue selector; A is built with mask-MULTIPLY (not select)
//   so the attn loads are unconditionally used and cannot be sunk into
//   exec-masked branch regions -> fully branchless inner loop.
// ---------------------------------------------------------------------------
__global__ void segsum_wmma_kernel(const float* __restrict__ feat,
                                   const int*   __restrict__ ridxp,   // padded
                                   const float* __restrict__ attn,
                                   float* __restrict__ out) {
    const int s0   = blockIdx.x * 16;           // NSEG = 16 * 3125 exactly
    const int tid  = threadIdx.x;
    const int wave = tid >> 5;                  // 8 waves
    const int lane = tid & 31;
    const int half = lane >> 4;                 // 0: K=0,1   1: K=2,3
    const int lm   = lane & 15;                 // M (for A) / N (for B,D)
    const int fb   = wave * 64;                 // this wave's 64 feature columns

    // Uniform per-block segment boundaries (same result in every lane)
    const int start = lower_bound_i32(ridxp, N_ATOMS, s0);
    const int end   = lower_bound_i32(ridxp, N_ATOMS, s0 + 16);

    v8f acc0 = {}, acc1 = {}, acc2 = {}, acc3 = {};

    for (int a = start; a < end; a += 4) {
        const int a0 = a + 2 * half;            // this lane-half's two atoms
        const int a1 = a0 + 1;
        const int c0 = min(a0, N_ATOMS - 1);    // branchless in-bounds clamp;
        const int c1 = min(a1, N_ATOMS - 1);    // mask zeroes these cases

        const int r0 = ridxp[a0];               // padded: always in-bounds
        const int r1 = ridxp[a1];

        // Row-selector masks as 0.0/1.0 (v_cndmask with inline constants);
        // multiplying keeps the attn loads unconditionally live -> no sinking
        // into saveexec regions.
        const float m0 = ((r0 - s0) == lm) ? 1.0f : 0.0f;
        const float m1 = ((r1 - s0) == lm) ? 1.0f : 0.0f;

        // A 16x4 f32: lanes 0-15 -> K=0,1 ; lanes 16-31 -> K=2,3 ; M = lane%16
        v2f A;
        A.x = attn[c0] * m0;
        A.y = attn[c1] * m1;

        // B 4x16 f32: lanes 0-15 -> K=0,1 ; lanes 16-31 -> K=2,3 ; N = lane%16
        const float* f0 = feat + (size_t)c0 * FEAT + fb + lm;
        const float* f1 = feat + (size_t)c1 * FEAT + fb + lm;
        __builtin_prefetch(f0 + 4 * FEAT, 0, 0);

        v2f B0, B1, B2, B3;
        B0.x = __builtin_nontemporal_load(f0);      B0.y = __builtin_nontemporal_load(f1);
        B1.x = __builtin_nontemporal_load(f0 + 16); B1.y = __builtin_nontemporal_load(f1 + 16);
        B2.x = __builtin_nontemporal_load(f0 + 32); B2.y = __builtin_nontemporal_load(f1 + 32);
        B3.x = __builtin_nontemporal_load(f0 + 48); B3.y = __builtin_nontemporal_load(f1 + 48);

        acc0 = __builtin_amdgcn_wmma_f32_16x16x4_f32(false, A, false, B0, (short)0, acc0, false, false);
        acc1 = __builtin_amdgcn_wmma_f32_16x16x4_f32(false, A, false, B1, (short)0, acc1, false, false);
        acc2 = __builtin_amdgcn_wmma_f32_16x16x4_f32(false, A, false, B2, (short)0, acc2, false, false);
        acc3 = __builtin_amdgcn_wmma_f32_16x16x4_f32(false, A, false, B3, (short)0, acc3, false, false);
    }

    // D layout: VGPR r -> M = r + 8*half, N = lane%16. One plain store per elem;
    // output (102 MB) stays L2-resident.
#pragma unroll
    for (int r = 0; r < 8; ++r) {
        const int m = r + half * 8;
        float* orow = out + (size_t)(s0 + m) * FEAT + fb + lm;
        orow[0]  = acc0[r];
        orow[16] = acc1[r];
        orow[32] = acc2[r];
        orow[48] = acc3[r];
    }
}

// ---------------------------------------------------------------------------
extern "C" void kernel_launch(void* const* d_in, const int* in_sizes, int n_in,
                              void* d_out, int out_size, void* d_ws, size_t ws_size,
                              hipStream_t stream) {
    const float* feat = (const float*)d_in[0];   // [N_ATOMS, FEAT]
    const int*   ridx = (const int*)d_in[1];     // [N_ATOMS] sorted
    const float* W    = (const float*)d_in[2];   // [FEAT, FEAT]
    // d_in[3] = proj_b : constant shift of all scores -> cancels in softmax
    const float* q    = (const float*)d_in[4];   // [FEAT]
    float* out = (float*)d_out;                  // [NSEG, FEAT]

    float* ws      = (float*)d_ws;
    float* wq      = ws;                         // 512
    float* scores  = wq + FEAT;                  // N_ATOMS (reused as attn)
    float* exp_s   = scores + N_ATOMS;           // N_ATOMS
    float* seg_max = exp_s + N_ATOMS;            // NSEG
    float* denom   = seg_max + NSEG;             // NSEG
    int*   ridx_pad = (int*)(denom + NSEG);      // N_ATOMS + 4  (~5.2 MB total)
    float* attn    = scores;                     // scores is dead after expsum

    wq_kernel    <<<2, 256, 0, stream>>>(W, q, wq);
    init_kernel  <<<(NSEG + 255) / 256, 256, 0, stream>>>(seg_max, denom);
    scores_kernel<<<N_ATOMS / 8, 256, 0, stream>>>(feat, ridx, wq, scores, seg_max);
    expsum_kernel<<<(N_ATOMS + 255) / 256, 256, 0, stream>>>(scores, ridx, seg_max, exp_s, denom);
    attn_kernel  <<<(N_ATOMS + 4 + 255) / 256, 256, 0, stream>>>(exp_s, ridx, denom, attn, ridx_pad);
    segsum_wmma_kernel<<<NSEG / 16, 256, 0, stream>>>(feat, ridx_pad, attn, out);
}